// STN_51771535785990
// MI455X (gfx1250) — compile-verified
//
#include <hip/hip_runtime.h>
#include <hip/hip_bf16.h>

// Shapes fixed by the reference setup_inputs(): x = [32, 3, 512, 512] fp32,
// scale_factors = [2] fp32. Output = [32, 3, 512, 512] fp32.
#define BATCH 32
#define CHAN  3
#define HDIM  512
#define WDIM  512
#define TPB   128   // 4 waves; each thread produces 4 consecutive output columns

typedef float v4f __attribute__((ext_vector_type(4)));   // native vector for NT stores

// ---------------------------------------------------------------------------
// Setup: precompute per-column and per-row bilinear taps once into d_ws.
//   ws[0      .. WDIM-1]     : {a0, a1, ix0c, ix1c}  (column weights / indices)
//   ws[WDIM .. WDIM+HDIM-1]  : {b0, b1, iy0c, iy1c}  (row weights / indices)
// Zeros-padding validity is folded into the weights (invalid tap -> weight 0).
// ---------------------------------------------------------------------------
__global__ __launch_bounds__(128)
void weights_setup_kernel(const float* __restrict__ scale, float4* __restrict__ ws)
{
    const int i = blockIdx.x * blockDim.x + threadIdx.x;
    if (i >= WDIM) return;                 // HDIM == WDIM == 512
    const float sx = scale[0];
    const float sy = scale[1];

    {   // column entry:  ix = 0.5*(sx*(2x+1-W) + (W-1))
        const float Wf = (float)WDIM;
        const float ix = 0.5f * (sx * (2.0f * (float)i + 1.0f - Wf) + (Wf - 1.0f));
        const float f  = floorf(ix);
        const float w1 = ix - f, w0 = 1.0f - w1;
        const int i0 = (int)f, i1 = i0 + 1;
        const float a0 = (i0 >= 0 && i0 < WDIM) ? w0 : 0.0f;
        const float a1 = (i1 >= 0 && i1 < WDIM) ? w1 : 0.0f;
        const int i0c = min(max(i0, 0), WDIM - 1);
        const int i1c = min(max(i1, 0), WDIM - 1);
        ws[i] = make_float4(a0, a1, __int_as_float(i0c), __int_as_float(i1c));
    }
    {   // row entry:  iy = 0.5*(sy*(2y+1-H) + (H-1))
        const float Hf = (float)HDIM;
        const float iy = 0.5f * (sy * (2.0f * (float)i + 1.0f - Hf) + (Hf - 1.0f));
        const float f  = floorf(iy);
        const float w1 = iy - f, w0 = 1.0f - w1;
        const int i0 = (int)f, i1 = i0 + 1;
        const float b0 = (i0 >= 0 && i0 < HDIM) ? w0 : 0.0f;
        const float b1 = (i1 >= 0 && i1 < HDIM) ? w1 : 0.0f;
        const int i0c = min(max(i0, 0), HDIM - 1);
        const int i1c = min(max(i1, 0), HDIM - 1);
        ws[WDIM + i] = make_float4(b0, b1, __int_as_float(i0c), __int_as_float(i1c));
    }
}

// ---------------------------------------------------------------------------
// Main kernel: one block per (plane, output row). Async-DMA the two source
// rows into LDS, fetch precomputed taps, gather + 6 FMA per pixel.
// ---------------------------------------------------------------------------
__global__ __launch_bounds__(TPB)
void diag_grid_sample_kernel(const float* __restrict__ xin,
                             const float4* __restrict__ colw,
                             const float4* __restrict__ roww,
                             float* __restrict__ out)
{
    const int y  = blockIdx.x;   // output row
    const int bc = blockIdx.y;   // fused batch*channel plane
    const int t  = threadIdx.x;

    // Uniform per-row taps (one cached b128 per block).
    const float4 rw = roww[y];
    const float b0 = rw.x, b1 = rw.y;
    const int iy0c = __float_as_int(rw.z);
    const int iy1c = __float_as_int(rw.w);

    __shared__ float row0[WDIM];
    __shared__ float row1[WDIM];

    const size_t plane = (size_t)bc * (size_t)(HDIM * WDIM);
    const float* g0 = xin + plane + (size_t)iy0c * WDIM + 4 * t;
    const float* g1 = xin + plane + (size_t)iy1c * WDIM + 4 * t;
    const unsigned l0 = (unsigned)(size_t)(&row0[4 * t]);
    const unsigned l1 = (unsigned)(size_t)(&row1[4 * t]);
    const unsigned long long ga0 = (unsigned long long)(size_t)g0;
    const unsigned long long ga1 = (unsigned long long)(size_t)g1;

    // gfx1250 async data mover: global -> LDS, 16B/lane, ASYNCcnt-tracked.
    asm volatile("global_load_async_to_lds_b128 %0, %1, off"
                 :: "v"(l0), "v"(ga0) : "memory");
    asm volatile("global_load_async_to_lds_b128 %0, %1, off"
                 :: "v"(l1), "v"(ga1) : "memory");

    // Overlap the DMA with the column-tap fetches (WGP$-resident table).
    float4 cw[4];
    #pragma unroll
    for (int j = 0; j < 4; ++j) cw[j] = colw[4 * t + j];

    asm volatile("s_wait_asynccnt 0" ::: "memory");
    __syncthreads();

    float res[4];
    #pragma unroll
    for (int j = 0; j < 4; ++j) {
        const int i0 = __float_as_int(cw[j].z);
        const int i1 = __float_as_int(cw[j].w);
        // row0/row1 are 2048B apart in LDS -> pairs into ds_load_2addr_stride64.
        const float v00 = row0[i0];
        const float v10 = row1[i0];
        const float v01 = row0[i1];
        const float v11 = row1[i1];
        const float xh0 = v00 * b0 + v10 * b1;   // H-interp first (ref order)
        const float xh1 = v01 * b0 + v11 * b1;
        res[j] = xh0 * cw[j].x + xh1 * cw[j].y;  // then W-interp
    }

    // Non-temporal 128-bit store: keep the input resident in the 192MB L2.
    v4f o = { res[0], res[1], res[2], res[3] };
    v4f* orow = (v4f*)(out + plane + (size_t)y * WDIM);
    __builtin_nontemporal_store(o, orow + t);
}

// ---------------------------------------------------------------------------
// Fallback (self-contained, computes weights inline) if d_ws is too small.
// ---------------------------------------------------------------------------
__global__ __launch_bounds__(TPB)
void diag_grid_sample_fallback(const float* __restrict__ xin,
                               const float* __restrict__ scale,
                               float* __restrict__ out)
{
    const int y  = blockIdx.x;
    const int bc = blockIdx.y;
    const int t  = threadIdx.x;

    const float sx = scale[0];
    const float sy = scale[1];

    const float Hf  = (float)HDIM;
    const float iy  = 0.5f * (sy * (2.0f * (float)y + 1.0f - Hf) + (Hf - 1.0f));
    const float iy0f = floorf(iy);
    const float wy1 = iy - iy0f;
    const float wy0 = 1.0f - wy1;
    const int   iy0 = (int)iy0f;
    const int   iy1 = iy0 + 1;
    const float b0  = (iy0 >= 0 && iy0 < HDIM) ? wy0 : 0.0f;
    const float b1  = (iy1 >= 0 && iy1 < HDIM) ? wy1 : 0.0f;
    const int   iy0c = min(max(iy0, 0), HDIM - 1);
    const int   iy1c = min(max(iy1, 0), HDIM - 1);

    __shared__ float row0[WDIM];
    __shared__ float row1[WDIM];

    const size_t plane = (size_t)bc * (size_t)(HDIM * WDIM);
    const float* g0 = xin + plane + (size_t)iy0c * WDIM + 4 * t;
    const float* g1 = xin + plane + (size_t)iy1c * WDIM + 4 * t;
    const unsigned l0 = (unsigned)(size_t)(&row0[4 * t]);
    const unsigned l1 = (unsigned)(size_t)(&row1[4 * t]);
    const unsigned long long ga0 = (unsigned long long)(size_t)g0;
    const unsigned long long ga1 = (unsigned long long)(size_t)g1;

    asm volatile("global_load_async_to_lds_b128 %0, %1, off"
                 :: "v"(l0), "v"(ga0) : "memory");
    asm volatile("global_load_async_to_lds_b128 %0, %1, off"
                 :: "v"(l1), "v"(ga1) : "memory");
    asm volatile("s_wait_asynccnt 0" ::: "memory");
    __syncthreads();

    const float Wf = (float)WDIM;
    float res[4];
    #pragma unroll
    for (int j = 0; j < 4; ++j) {
        const int   xc   = 4 * t + j;
        const float ix   = 0.5f * (sx * (2.0f * (float)xc + 1.0f - Wf) + (Wf - 1.0f));
        const float ix0f = floorf(ix);
        const float wx1  = ix - ix0f;
        const float wx0  = 1.0f - wx1;
        const int   ix0  = (int)ix0f;
        const int   ix1  = ix0 + 1;
        const float a0   = (ix0 >= 0 && ix0 < WDIM) ? wx0 : 0.0f;
        const float a1   = (ix1 >= 0 && ix1 < WDIM) ? wx1 : 0.0f;
        const int   ix0c = min(max(ix0, 0), WDIM - 1);
        const int   ix1c = min(max(ix1, 0), WDIM - 1);
        const float xh0 = row0[ix0c] * b0 + row1[ix0c] * b1;
        const float xh1 = row0[ix1c] * b0 + row1[ix1c] * b1;
        res[j] = xh0 * a0 + xh1 * a1;
    }

    v4f o = { res[0], res[1], res[2], res[3] };
    v4f* orow = (v4f*)(out + plane + (size_t)y * WDIM);
    __builtin_nontemporal_store(o, orow + t);
}

extern "C" void kernel_launch(void* const* d_in, const int* in_sizes, int n_in,
                              void* d_out, int out_size, void* d_ws, size_t ws_size,
                              hipStream_t stream) {
    (void)in_sizes; (void)n_in; (void)out_size;
    const float* xin   = (const float*)d_in[0];
    const float* scale = (const float*)d_in[1];
    float* outp        = (float*)d_out;

    const size_t need = (size_t)(WDIM + HDIM) * sizeof(float4);   // 16 KB
    dim3 grid(HDIM, BATCH * CHAN);

    if (ws_size >= need && d_ws != nullptr) {
        float4* tab = (float4*)d_ws;
        weights_setup_kernel<<<(WDIM + 127) / 128, 128, 0, stream>>>(scale, tab);
        diag_grid_sample_kernel<<<grid, TPB, 0, stream>>>(xin, tab, tab + WDIM, outp);
    } else {
        diag_grid_sample_fallback<<<grid, TPB, 0, stream>>>(xin, scale, outp);
    }
}